// SubstructureLayer_44744969290501
// MI455X (gfx1250) — compile-verified
//
#include <hip/hip_runtime.h>

#define N_NODES 50000
#define N_SUB   20000
#define N_EDGES 600000
#define E_SUB   120000
#define DFEAT   128
#define LSTR    132     // padded LDS row stride (floats): bank = (4n+k)%64 -> conflict-free b64 loads

typedef __attribute__((ext_vector_type(2))) float v2f;
typedef __attribute__((ext_vector_type(4))) float v4f;
typedef __attribute__((ext_vector_type(8))) float v8f;

// ---------------------------------------------------------------------------
// zero-fill (float4)
// ---------------------------------------------------------------------------
__global__ void zero_kernel(v4f* __restrict__ p, int n4) {
    int i = blockIdx.x * blockDim.x + threadIdx.x;
    if (i < n4) p[i] = (v4f){0.f, 0.f, 0.f, 0.f};
}

// ---------------------------------------------------------------------------
// copy (float4): dst = src   (used to seed agg buffer with x)
// ---------------------------------------------------------------------------
__global__ void copy_kernel(const v4f* __restrict__ src, v4f* __restrict__ dst, int n4) {
    int i = blockIdx.x * blockDim.x + threadIdx.x;
    if (i < n4) dst[i] = src[i];
}

// ---------------------------------------------------------------------------
// scatter-add: out[dst[e]][:] += feat[src[e]][:]   (hardware f32 atomics)
// 128 threads per edge, 2 edges per 256-thread block
// ---------------------------------------------------------------------------
__global__ __launch_bounds__(256)
void scatter_add_kernel(const float* __restrict__ feat,
                        const int* __restrict__ src_idx,
                        const int* __restrict__ dst_idx,
                        float* __restrict__ out, int n_edges) {
    int e = blockIdx.x * 2 + (threadIdx.x >> 7);
    int d = threadIdx.x & 127;
    if (e >= n_edges) return;
    int s = src_idx[e];
    int t = dst_idx[e];
    float v = feat[(size_t)s * DFEAT + d];
    __hip_atomic_fetch_add(&out[(size_t)t * DFEAT + d], v,
                           __ATOMIC_RELAXED, __HIP_MEMORY_SCOPE_AGENT);
}

// ---------------------------------------------------------------------------
// Y[M,128] = act( A @ W + bias [+ R] )
// One 256-thread block (8 waves) per 128 output rows.
// Wave computes a 16x128 stripe with 8 f32 accumulator tiles via
// V_WMMA_F32_16X16X4_F32 (exact fp32), K-loop of 32 steps.
// W^T staged in dynamic LDS with padded stride 132: B-fragment ds_load_b64
// addresses are base + compile-time immediates (no VALU address math),
// and the 64 dwords per wave-access hit all 64 LDS banks (conflict-free).
// ---------------------------------------------------------------------------
template <bool HAS_R, bool RELU>
__global__ __launch_bounds__(256)
void gemm128_kernel(const float* __restrict__ A,
                    const float* __restrict__ W,      // [128,128] row-major
                    const float* __restrict__ bias,   // [128]
                    const float* __restrict__ R,      // residual [M,128] (if HAS_R)
                    float* __restrict__ Y,
                    int M) {
    extern __shared__ float WT[];                     // [128][LSTR] = 67.6 KB, W transposed

    int tid = threadIdx.x;
    // stage W transposed into LDS (coalesced global reads)
    for (int i = tid; i < DFEAT * DFEAT; i += 256) {
        int k = i >> 7;
        int n = i & 127;
        WT[n * LSTR + k] = W[i];
    }
    __syncthreads();

    int wave = tid >> 5;
    int lane = tid & 31;
    int m0 = blockIdx.x * 128 + wave * 16;
    if (m0 >= M) return;                              // uniform per wave: EXEC stays all-ones

    int r  = lane & 15;                               // row within A-tile / col within B-tile
    int kh = (lane >> 4) << 1;                        // lane-group K offset: 0 or 2

    const float* arow  = A + (size_t)(m0 + r) * DFEAT;
    const float* wbase = &WT[r * LSTR + kh];          // LDS base; nt offsets are immediates

    v8f acc[8] = {};

    for (int k0 = 0; k0 < DFEAT; k0 += 4) {
        v2f a = *reinterpret_cast<const v2f*>(&arow[k0 + kh]);   // global_load_b64

        // 8 independent ds_load_b64, all immediate-offset off one base
        v2f b[8];
#pragma unroll
        for (int nt = 0; nt < 8; ++nt)
            b[nt] = *reinterpret_cast<const v2f*>(&wbase[nt * 16 * LSTR + k0]);

#pragma unroll
        for (int nt = 0; nt < 8; ++nt)
            acc[nt] = __builtin_amdgcn_wmma_f32_16x16x4_f32(
                false, a, false, b[nt], (short)0, acc[nt], false, false);
    }

    // D layout: VGPR j -> row m0 + j + 8*(lane>=16); col = nt*16 + (lane&15)
    int rbase = m0 + ((lane >> 4) << 3);
#pragma unroll
    for (int nt = 0; nt < 8; ++nt) {
        int n = nt * 16 + r;
        float bv = bias[n];
#pragma unroll
        for (int j = 0; j < 8; ++j) {
            size_t idx = (size_t)(rbase + j) * DFEAT + n;
            float v = acc[nt][j] + bv;
            if (HAS_R) v += R[idx];
            if (RELU)  v = v > 0.0f ? v : 0.0f;
            Y[idx] = v;
        }
    }
}

// ---------------------------------------------------------------------------
// launch helpers
// ---------------------------------------------------------------------------
static inline void launch_zero(float* p, int n, hipStream_t s) {
    int n4 = n >> 2;
    zero_kernel<<<(n4 + 1023) / 1024, 1024, 0, s>>>((v4f*)p, n4);
}
static inline void launch_copy(const float* src, float* dst, int n, hipStream_t s) {
    int n4 = n >> 2;
    copy_kernel<<<(n4 + 1023) / 1024, 1024, 0, s>>>((const v4f*)src, (v4f*)dst, n4);
}
static inline void launch_scatter(const float* feat, const int* si, const int* di,
                                  float* out, int ne, hipStream_t s) {
    scatter_add_kernel<<<(ne + 1) / 2, 256, 0, s>>>(feat, si, di, out, ne);
}
template <bool HAS_R, bool RELU>
static inline void launch_gemm(const float* A, const float* W, const float* b,
                               const float* R, float* Y, int M, hipStream_t s) {
    size_t lds_bytes = (size_t)DFEAT * LSTR * sizeof(float);   // 67584
    gemm128_kernel<HAS_R, RELU><<<(M + 127) / 128, 256, lds_bytes, s>>>(A, W, b, R, Y, M);
}

extern "C" void kernel_launch(void* const* d_in, const int* in_sizes, int n_in,
                              void* d_out, int out_size, void* d_ws, size_t ws_size,
                              hipStream_t stream) {
    const float* x     = (const float*)d_in[0];
    const float* Wm    = (const float*)d_in[1];
    const float* bm    = (const float*)d_in[2];
    const float* Wn2s0 = (const float*)d_in[3];
    const float* bn2s0 = (const float*)d_in[4];
    const float* Ws2n0 = (const float*)d_in[5];
    const float* bs2n0 = (const float*)d_in[6];
    const float* Wn2s1 = (const float*)d_in[7];
    const float* bn2s1 = (const float*)d_in[8];
    const float* Ws2n1 = (const float*)d_in[9];
    const float* bs2n1 = (const float*)d_in[10];
    const int*   edges = (const int*)d_in[11];      // [2, N_EDGES]
    const int*   row0  = (const int*)d_in[12];
    const int*   col0  = (const int*)d_in[13];
    const int*   row1  = (const int*)d_in[14];
    const int*   col1  = (const int*)d_in[15];
    float* out = (float*)d_out;

    const int ND = N_NODES * DFEAT;                 // 6.4M floats
    const int SD = N_SUB * DFEAT;                   // 2.56M floats

    float* buf0 = (float*)d_ws;                     // agg / msg scratch  [N,D]
    float* buf1 = buf0 + (size_t)ND;                // h (post-relu x)    [N,D]
    float* buf4 = buf1 + (size_t)ND;                // h after level 0    [N,D]
    float* buf2 = buf4 + (size_t)ND;                // sub scatter raw    [S,D]
    float* buf3 = buf2 + (size_t)SD;                // sub after gemm     [S,D]

    const int* esrc = edges;
    const int* edst = edges + N_EDGES;

    // ---- neighbor aggregation: buf0 = x, buf0[dst] += x[src]; h = relu(buf0@Wm+bm)
    launch_copy(x, buf0, ND, stream);
    launch_scatter(x, esrc, edst, buf0, N_EDGES, stream);
    launch_gemm<false, true>(buf0, Wm, bm, nullptr, buf1, N_NODES, stream);

    // ---- level 0 ----
    launch_zero(buf2, SD, stream);
    launch_scatter(buf1, row0, col0, buf2, E_SUB, stream);       // node -> sub
    launch_gemm<false, false>(buf2, Wn2s0, bn2s0, nullptr, buf3, N_SUB, stream);
    launch_zero(buf0, ND, stream);
    launch_scatter(buf3, col0, row0, buf0, E_SUB, stream);       // sub -> node
    launch_gemm<true, false>(buf0, Ws2n0, bs2n0, buf1, buf4, N_NODES, stream);

    // ---- level 1 ----
    launch_zero(buf2, SD, stream);
    launch_scatter(buf4, row1, col1, buf2, E_SUB, stream);
    launch_gemm<false, false>(buf2, Wn2s1, bn2s1, nullptr, buf3, N_SUB, stream);
    launch_zero(buf0, ND, stream);
    launch_scatter(buf3, col1, row1, buf0, E_SUB, stream);
    launch_gemm<true, false>(buf0, Ws2n1, bs2n1, buf4, out, N_NODES, stream);
}